// UnweightedNeRF_26079041421676
// MI455X (gfx1250) — compile-verified
//
#include <hip/hip_runtime.h>

typedef __attribute__((ext_vector_type(2))) float v2f;
typedef __attribute__((ext_vector_type(8))) float v8f;

#define HDIM 64
#define EDIM 128
#define LDIM 4
#define LSTR 68  // LDS row stride in floats (64 + 4 pad -> conflict-free)

__global__ __launch_bounds__(32) void moe_mlp16_f32_wmma(
    const float* __restrict__ x,        // [N, 64]
    const float* __restrict__ weights,  // [L, E, 64, 64] (out, in)
    const float* __restrict__ biases,   // [L, E, 64]
    const long long* __restrict__ index,// [N] int64, sorted
    float* __restrict__ out,            // [N, 64]
    int N)
{
    __shared__ float lds[16 * LSTR];

    const int lane = threadIdx.x;      // 0..31 (wave32)
    const int l16  = lane & 15;
    const int half = lane >> 4;        // 0 or 1
    const int base = blockIdx.x * 16;  // first sample row of this tile
    const int rowsValid = (N - base) < 16 ? (N - base) : 16;

    // ---- stage 16 rows of x into LDS (coalesced float2, conflict-free) ----
    #pragma unroll
    for (int i = 0; i < 16; ++i) {
        v2f v = {0.0f, 0.0f};
        if (i < rowsValid)
            v = *(const v2f*)(x + ((size_t)(base + i)) * HDIM + 2 * lane);
        *(v2f*)&lds[i * LSTR + 2 * lane] = v;
    }
    asm volatile("s_wait_dscnt 0" ::: "memory");

    // expert id of each tile row (lanes 0..15 meaningful; 16..31 mirror)
    int eidx = base + l16;
    if (eidx >= N) eidx = N - 1;
    const int e_row = (int)index[eidx];

    for (int l = 0; l < LDIM; ++l) {
        // ---- load A fragments from LDS ----
        // A 16x4 f32 layout: lane=M, VGPR0/1 = K {2*half, 2*half+1} of chunk k
        v2f a[16];
        #pragma unroll
        for (int k = 0; k < 16; ++k)
            a[k] = *(const v2f*)&lds[l16 * LSTR + 4 * k + 2 * half];
        asm volatile("s_wait_dscnt 0" ::: "memory");

        // ---- uniform loop over distinct experts in this 16-row tile ----
        unsigned pending = 0xFFFFu;
        while (pending) {
            const int r = (int)__builtin_ctz(pending);
            int e = __shfl(e_row, r, 32);                    // expert of row r
            e = __builtin_amdgcn_readfirstlane(e);           // make scalar
            const unsigned match =
                (unsigned)__ballot(lane < 16 && e_row == e) & 0xFFFFu; // uniform
            pending &= ~match;

            const float* __restrict__ W =
                weights + (((size_t)l * EDIM + e) * HDIM) * HDIM;
            const float* __restrict__ B =
                biases + ((size_t)l * EDIM + e) * HDIM;

            // out[m][n] = sum_k h[m][k] * W[n][k]  (B fragment = W[n][kg..kg+1])
            v8f c0 = {}, c1 = {}, c2 = {}, c3 = {};
            #pragma unroll
            for (int k = 0; k < 16; ++k) {
                const int kg = 4 * k + 2 * half;
                const v2f b0 = *(const v2f*)&W[(size_t)( 0 + l16) * HDIM + kg];
                const v2f b1 = *(const v2f*)&W[(size_t)(16 + l16) * HDIM + kg];
                const v2f b2 = *(const v2f*)&W[(size_t)(32 + l16) * HDIM + kg];
                const v2f b3 = *(const v2f*)&W[(size_t)(48 + l16) * HDIM + kg];
                c0 = __builtin_amdgcn_wmma_f32_16x16x4_f32(false, a[k], false, b0, (short)0, c0, false, false);
                c1 = __builtin_amdgcn_wmma_f32_16x16x4_f32(false, a[k], false, b1, (short)0, c1, false, false);
                c2 = __builtin_amdgcn_wmma_f32_16x16x4_f32(false, a[k], false, b2, (short)0, c2, false, false);
                c3 = __builtin_amdgcn_wmma_f32_16x16x4_f32(false, a[k], false, b3, (short)0, c3, false, false);
            }

            const float bv0 = B[ 0 + l16];
            const float bv1 = B[16 + l16];
            const float bv2 = B[32 + l16];
            const float bv3 = B[48 + l16];

            // merge rows whose expert == e back into LDS (bias + relu)
            // C/D layout: VGPR j -> row (j + 8*half), col = l16 within n-tile
            #pragma unroll
            for (int j = 0; j < 8; ++j) {
                const int m = j + half * 8;
                if ((match >> m) & 1u) {
                    float t0 = c0[j] + bv0;
                    float t1 = c1[j] + bv1;
                    float t2 = c2[j] + bv2;
                    float t3 = c3[j] + bv3;
                    lds[m * LSTR +  0 + l16] = t0 > 0.0f ? t0 : 0.0f;
                    lds[m * LSTR + 16 + l16] = t1 > 0.0f ? t1 : 0.0f;
                    lds[m * LSTR + 32 + l16] = t2 > 0.0f ? t2 : 0.0f;
                    lds[m * LSTR + 48 + l16] = t3 > 0.0f ? t3 : 0.0f;
                }
            }
        }
        asm volatile("s_wait_dscnt 0" ::: "memory");
    }

    // ---- write result tile back to global (coalesced float2) ----
    #pragma unroll
    for (int i = 0; i < 16; ++i) {
        if (i < rowsValid) {
            v2f v = *(const v2f*)&lds[i * LSTR + 2 * lane];
            *(v2f*)(out + ((size_t)(base + i)) * HDIM + 2 * lane) = v;
        }
    }
}

extern "C" void kernel_launch(void* const* d_in, const int* in_sizes, int n_in,
                              void* d_out, int out_size, void* d_ws, size_t ws_size,
                              hipStream_t stream) {
    const float*     x   = (const float*)d_in[0];     // [N,64] f32
    const float*     w   = (const float*)d_in[1];     // [4,128,64,64] f32
    const float*     b   = (const float*)d_in[2];     // [4,128,64] f32
    const long long* idx = (const long long*)d_in[3]; // [N] int64 (sorted)
    float*           out = (float*)d_out;             // [N,64] f32

    const int N = in_sizes[0] / HDIM;
    const int tiles = (N + 15) / 16;
    moe_mlp16_f32_wmma<<<tiles, 32, 0, stream>>>(x, w, b, idx, out, N);
}